// DGCNN_20710332301414
// MI455X (gfx1250) — compile-verified
//
#include <hip/hip_runtime.h>
#include <math.h>

// DGCNN forward for MI455X (gfx1250, wave32).
// All GEMM-shaped work uses V_WMMA_F32_16X16X4_F32 (fp32 matches reference).
// LDS tiles are padded (+4 floats/row) so WMMA A-fragment loads hit 16
// distinct banks. Top-K is wave-local (wave32 shfl_xor reductions, no block
// barriers in the selection loop). Workspace usage: ~104 MB.

typedef float v2f __attribute__((ext_vector_type(2)));
typedef float v8f __attribute__((ext_vector_type(8)));

static constexpr int BB   = 8;     // batch
static constexpr int NPTS = 2048;  // points per cloud
static constexpr int KNN  = 20;    // neighbors
static constexpr float NEGINF = -3.4e38f;

__device__ __forceinline__ v8f wmma_f32(v2f a, v2f b, v8f c) {
  // 8 args: (neg_a, A, neg_b, B, c_mod, C, reuse_a, reuse_b)
  return __builtin_amdgcn_wmma_f32_16x16x4_f32(false, a, false, b, (short)0, c,
                                               false, false);
}

// ---------------------------------------------------------------------------
// Squared norms per point: d2[b*N+n] = sum_c x^2
// ---------------------------------------------------------------------------
__global__ void d2_kernel(const float* __restrict__ x, int ld, int off, int C,
                          float* __restrict__ d2) {
  int i = blockIdx.x * blockDim.x + threadIdx.x;
  if (i >= BB * NPTS) return;
  const float* p = x + (size_t)i * ld + off;
  float s = 0.f;
  for (int c = 0; c < C; ++c) { float v = p[c]; s += v * v; }
  d2[i] = s;
}

// ---------------------------------------------------------------------------
// kNN: for a tile of 16 points, build pd row (2<xi,xj> - |xi|^2 - |xj|^2) via
// WMMA f32 Gram tiles into LDS, then wave-local 20x argmax selection per row
// (each of the 8 waves owns 2 rows; per-wave LDS ops are in-order, so the
// lane-0 "-inf" poison needs no block barrier).
// ---------------------------------------------------------------------------
template <int C>
__global__ __launch_bounds__(256) void knn_kernel(
    const float* __restrict__ x, int ld, int off,
    const float* __restrict__ d2, int* __restrict__ knnidx) {
  constexpr int CP  = (C + 3) & ~3;
  constexpr int SRP = CP + 4;               // padded row stride (bank-spread)
  __shared__ float sdist[16][NPTS];         // 128 KB (CDNA5: up to 320 KB/WG)
  __shared__ float srow[16][SRP];
  __shared__ float sd2r[16];

  const int tid = threadIdx.x, lane = tid & 31, wave = tid >> 5;
  const int gp0 = blockIdx.x * 16;          // global point base (same b)
  const int bbase = (gp0 / NPTS) * NPTS;

  for (int i = tid; i < 16 * CP; i += 256) {
    int p = i / CP, c = i % CP;
    srow[p][c] = (c < C) ? x[(size_t)(gp0 + p) * ld + off + c] : 0.f;
  }
  if (tid < 16) sd2r[tid] = d2[gp0 + tid];
  __syncthreads();

  const int m  = lane & 15;
  const int kh = (lane >> 4) * 2;
  for (int j = wave; j < NPTS / 16; j += 8) {
    const int cb = j * 16;
    const int nn = cb + m;                  // this lane's output column
    const float* colp = x + (size_t)(bbase + nn) * ld + off;
    v8f acc = {0.f, 0.f, 0.f, 0.f, 0.f, 0.f, 0.f, 0.f};
#pragma unroll
    for (int k0 = 0; k0 < CP; k0 += 4) {
      v2f a, b;
      a.x = srow[m][k0 + kh];
      a.y = srow[m][k0 + kh + 1];
      int kc = k0 + kh;
      b.x = (kc     < C) ? colp[kc]     : 0.f;
      b.y = (kc + 1 < C) ? colp[kc + 1] : 0.f;
      acc = wmma_f32(a, b, acc);
    }
    float d2c = d2[bbase + nn];
#pragma unroll
    for (int r = 0; r < 8; ++r) {
      int mr = r + 8 * (lane >> 4);
      sdist[mr][nn] = 2.f * acc[r] - sd2r[mr] - d2c;
    }
  }
  __syncthreads();

  // Wave-local top-K (matches jax top_k: descending values, ties -> lower idx)
  for (int p = wave; p < 16; p += 8) {
    for (int kq = 0; kq < KNN; ++kq) {
      float best = NEGINF; int bi = NPTS;
      for (int mc = lane; mc < NPTS; mc += 32) {
        float v = sdist[p][mc];
        if (v > best || (v == best && mc < bi)) { best = v; bi = mc; }
      }
#pragma unroll
      for (int o = 16; o > 0; o >>= 1) {
        float vv = __shfl_xor(best, o, 32);
        int   ii = __shfl_xor(bi,   o, 32);
        if (vv > best || (vv == best && ii < bi)) { best = vv; bi = ii; }
      }
      if (lane == 0) {
        knnidx[(size_t)(gp0 + p) * KNN + kq] = bi;
        sdist[p][bi] = NEGINF;  // in-order per-wave LDS: visible to next scan
      }
    }
  }
}

// ---------------------------------------------------------------------------
// EdgeConv: per 16-point tile, for each k gather edge features into LDS,
// WMMA [16 x 2C] x [2C x COUT], BN + LeakyReLU, running max over k in VGPRs.
// W fragments are k-invariant: preloaded into VGPRs when small enough.
// ---------------------------------------------------------------------------
template <int CIN, int COUT, int NW>
__global__ __launch_bounds__(NW * 32) void edgeconv_kernel(
    const float* __restrict__ x, int ld_in, int off_in,
    const int* __restrict__ knnidx,
    const float* __restrict__ W,      // [COUT][2*CIN] row-major
    const float* __restrict__ bnp,    // [4][COUT]: gamma, beta, mean, var
    float* __restrict__ out, int ld_out, int off_out) {
  constexpr int C2     = 2 * CIN;
  constexpr int C2P    = (C2 + 3) & ~3;
  constexpr int LDA    = C2P + 4;          // padded row stride (bank-spread)
  constexpr int NT     = COUT / 16;
  constexpr int TPW    = NT / NW;
  constexpr int KSTEPS = C2P / 4;
  constexpr bool PRE   = (TPW * KSTEPS) <= 32;  // preload B frags into VGPRs

  __shared__ float sA[16][LDA];   // edge features for one k (M=16 points)
  __shared__ float sC[16][CIN];   // center features
  __shared__ int   sI[16];

  const int tid = threadIdx.x, lane = tid & 31, wave = tid >> 5;
  const int gp0 = blockIdx.x * 16;
  const int bbase = (gp0 / NPTS) * NPTS;
  const int m  = lane & 15;
  const int kh = (lane >> 4) * 2;

  for (int i = tid; i < 16 * CIN; i += NW * 32) {
    int p = i / CIN, c = i % CIN;
    sC[p][c] = x[(size_t)(gp0 + p) * ld_in + off_in + c];
  }
  // zero-pad the K-dimension tail once (never rewritten)
  constexpr int PAD = (C2P > C2) ? (C2P - C2) : 1;
  for (int i = tid; i < 16 * (C2P - C2); i += NW * 32) {
    int p = i / PAD, c = i % PAD;
    sA[p][C2 + c] = 0.f;
  }

  // per-lane fused BN scale/shift for each owned output column
  float scl[TPW], sft[TPW];
#pragma unroll
  for (int t = 0; t < TPW; ++t) {
    int o = (wave + t * NW) * 16 + m;
    float g = bnp[o], b = bnp[COUT + o], mm = bnp[2 * COUT + o], vv = bnp[3 * COUT + o];
    float s = g / sqrtf(vv + 1e-5f);
    scl[t] = s; sft[t] = b - mm * s;
  }

  // preload k-invariant W fragments (B column = W row nb+m)
  v2f breg[PRE ? TPW * KSTEPS : 1];
  if (PRE) {
#pragma unroll
    for (int t = 0; t < TPW; ++t) {
      const float* wr = W + (size_t)((wave + t * NW) * 16 + m) * C2;
#pragma unroll
      for (int ks = 0; ks < KSTEPS; ++ks) {
        int kc = ks * 4 + kh;
        breg[t * KSTEPS + ks].x = (kc     < C2) ? wr[kc]     : 0.f;
        breg[t * KSTEPS + ks].y = (kc + 1 < C2) ? wr[kc + 1] : 0.f;
      }
    }
  }

  v8f vmax[TPW];
#pragma unroll
  for (int t = 0; t < TPW; ++t)
#pragma unroll
    for (int r = 0; r < 8; ++r) vmax[t][r] = NEGINF;

  for (int k = 0; k < KNN; ++k) {
    __syncthreads();
    if (tid < 16) sI[tid] = knnidx[(size_t)(gp0 + tid) * KNN + k];
    __syncthreads();
    for (int i = tid; i < 16 * C2; i += NW * 32) {
      int p = i / C2, c = i % C2;
      float ctr = sC[p][(c < CIN) ? c : (c - CIN)];
      float v;
      if (c < CIN) {
        const float* np = x + (size_t)(bbase + sI[p]) * ld_in + off_in;
        v = np[c] - ctr;                  // neighbor - center
      } else {
        v = ctr;                           // center
      }
      sA[p][c] = v;
    }
    __syncthreads();
#pragma unroll
    for (int t = 0; t < TPW; ++t) {
      const int nb = (wave + t * NW) * 16;
      const float* wr = W + (size_t)(nb + m) * C2;
      v8f acc = {0.f, 0.f, 0.f, 0.f, 0.f, 0.f, 0.f, 0.f};
#pragma unroll
      for (int ks = 0; ks < KSTEPS; ++ks) {
        const int k0 = ks * 4;
        v2f a, b;
        a.x = sA[m][k0 + kh];
        a.y = sA[m][k0 + kh + 1];
        if (PRE) {
          b = breg[t * KSTEPS + ks];
        } else {
          int kc = k0 + kh;
          b.x = (kc     < C2) ? wr[kc]     : 0.f;
          b.y = (kc + 1 < C2) ? wr[kc + 1] : 0.f;
        }
        acc = wmma_f32(a, b, acc);
      }
#pragma unroll
      for (int r = 0; r < 8; ++r) {
        float h = acc[r] * scl[t] + sft[t];
        h = (h > 0.f) ? h : 0.2f * h;
        if (h > vmax[t][r]) vmax[t][r] = h;
      }
    }
  }
#pragma unroll
  for (int t = 0; t < TPW; ++t) {
    const int nb = (wave + t * NW) * 16;
#pragma unroll
    for (int r = 0; r < 8; ++r) {
      int p = r + 8 * (lane >> 4);
      out[(size_t)(gp0 + p) * ld_out + off_out + nb + m] = vmax[t][r];
    }
  }
}

// ---------------------------------------------------------------------------
// Pointwise GEMM + BN + LeakyReLU: out[16 x COUT] = act(bn(in[16 x CIN] @ W^T))
// ---------------------------------------------------------------------------
template <int CIN, int COUT, int NW>
__global__ __launch_bounds__(NW * 32) void rowgemm_bn_act_kernel(
    const float* __restrict__ x, int ld_in, int off_in,
    const float* __restrict__ W,      // [COUT][CIN]
    const float* __restrict__ bnp,    // [4][COUT]
    float* __restrict__ out, int ld_out) {
  constexpr int NT = COUT / 16, TPW = NT / NW;
  constexpr int LDA = CIN + 4;            // padded row stride (bank-spread)
  __shared__ float sA[16][LDA];
  const int tid = threadIdx.x, lane = tid & 31, wave = tid >> 5;
  const int gp0 = blockIdx.x * 16;
  for (int i = tid; i < 16 * CIN; i += NW * 32) {
    int p = i / CIN, c = i % CIN;
    sA[p][c] = x[(size_t)(gp0 + p) * ld_in + off_in + c];
  }
  __syncthreads();
  const int m  = lane & 15;
  const int kh = (lane >> 4) * 2;
#pragma unroll
  for (int t = 0; t < TPW; ++t) {
    const int nb = (wave + t * NW) * 16;
    const int o  = nb + m;
    float g = bnp[o], b = bnp[COUT + o], mm = bnp[2 * COUT + o], vv = bnp[3 * COUT + o];
    float s = g / sqrtf(vv + 1e-5f);
    float sh = b - mm * s;
    const float* wr = W + (size_t)o * CIN;
    v8f acc = {0.f, 0.f, 0.f, 0.f, 0.f, 0.f, 0.f, 0.f};
    for (int k0 = 0; k0 < CIN; k0 += 4) {
      v2f a, bf;
      a.x  = sA[m][k0 + kh];
      a.y  = sA[m][k0 + kh + 1];
      bf.x = wr[k0 + kh];
      bf.y = wr[k0 + kh + 1];
      acc = wmma_f32(a, bf, acc);
    }
#pragma unroll
    for (int r = 0; r < 8; ++r) {
      float h = acc[r] * s + sh;
      h = (h > 0.f) ? h : 0.2f * h;
      int p = r + 8 * (lane >> 4);
      out[(size_t)(gp0 + p) * ld_out + nb + m] = h;
    }
  }
}

// ---------------------------------------------------------------------------
// Global max + mean pooling over N, two stages.
// ---------------------------------------------------------------------------
__global__ void pool1_kernel(const float* __restrict__ x5,
                             float* __restrict__ pmax, float* __restrict__ psum) {
  const int b = blockIdx.y, seg = blockIdx.x, tid = threadIdx.x;
  const int C = 1024, SEGN = NPTS / 16;
  float mx[4], sm[4];
#pragma unroll
  for (int q = 0; q < 4; ++q) { mx[q] = NEGINF; sm[q] = 0.f; }
  const float* base = x5 + ((size_t)b * NPTS + (size_t)seg * SEGN) * C;
  for (int n = 0; n < SEGN; ++n) {
#pragma unroll
    for (int q = 0; q < 4; ++q) {
      float v = base[(size_t)n * C + q * 256 + tid];
      if (v > mx[q]) mx[q] = v;
      sm[q] += v;
    }
  }
#pragma unroll
  for (int q = 0; q < 4; ++q) {
    pmax[((size_t)b * 16 + seg) * C + q * 256 + tid] = mx[q];
    psum[((size_t)b * 16 + seg) * C + q * 256 + tid] = sm[q];
  }
}

__global__ void pool2_kernel(const float* __restrict__ pmax,
                             const float* __restrict__ psum, float* __restrict__ g) {
  int i = blockIdx.x * blockDim.x + threadIdx.x;
  if (i >= BB * 1024) return;
  int b = i / 1024, o = i % 1024;
  float mx = NEGINF, sm = 0.f;
  for (int s = 0; s < 16; ++s) {
    float v = pmax[((size_t)b * 16 + s) * 1024 + o];
    if (v > mx) mx = v;
    sm += psum[((size_t)b * 16 + s) * 1024 + o];
  }
  g[(size_t)b * 2048 + o]        = mx;
  g[(size_t)b * 2048 + 1024 + o] = sm * (1.f / 2048.f);
}

// ---------------------------------------------------------------------------
// Tiny FC head (B=8 rows): one thread per (b, o).
// ---------------------------------------------------------------------------
__global__ void head_fc_kernel(const float* __restrict__ in, const float* __restrict__ W,
                               const float* __restrict__ bias, const float* __restrict__ bnp,
                               float* __restrict__ out, int CI, int CO, int doact) {
  int i = blockIdx.x * blockDim.x + threadIdx.x;
  if (i >= BB * CO) return;
  int b = i / CO, o = i % CO;
  const float* ip = in + (size_t)b * CI;
  const float* wr = W + (size_t)o * CI;
  float acc = bias ? bias[o] : 0.f;
  for (int c = 0; c < CI; ++c) acc += ip[c] * wr[c];
  if (bnp) {
    float g = bnp[o], bb = bnp[CO + o], mm = bnp[2 * CO + o], vv = bnp[3 * CO + o];
    float s = g / sqrtf(vv + 1e-5f);
    acc = acc * s + (bb - mm * s);
  }
  if (doact) acc = (acc > 0.f) ? acc : 0.2f * acc;
  out[i] = acc;
}

// ---------------------------------------------------------------------------
extern "C" void kernel_launch(void* const* d_in, const int* in_sizes, int n_in,
                              void* d_out, int out_size, void* d_ws, size_t ws_size,
                              hipStream_t stream) {
  (void)in_sizes; (void)n_in; (void)out_size; (void)ws_size;
  const float* x   = (const float*)d_in[0];
  // d_in[1] = normal (unused on this path)
  const float* w1  = (const float*)d_in[2];
  const float* w2  = (const float*)d_in[3];
  const float* w3  = (const float*)d_in[4];
  const float* w4  = (const float*)d_in[5];
  const float* w5  = (const float*)d_in[6];
  const float* wl1 = (const float*)d_in[7];
  const float* wl2 = (const float*)d_in[8];
  const float* bl2 = (const float*)d_in[9];
  const float* wl3 = (const float*)d_in[10];
  const float* bl3 = (const float*)d_in[11];
  const float* bn1 = (const float*)d_in[12];
  const float* bn2 = (const float*)d_in[13];
  const float* bn3 = (const float*)d_in[14];
  const float* bn4 = (const float*)d_in[15];
  const float* bn5 = (const float*)d_in[16];
  const float* bn6 = (const float*)d_in[17];
  const float* bn7 = (const float*)d_in[18];
  float* out = (float*)d_out;

  char* ws = (char*)d_ws;
  auto alloc = [&](size_t bytes) -> char* {
    char* p = ws; ws += (bytes + 255) & ~(size_t)255; return p;
  };
  float* xc   = (float*)alloc((size_t)BB * NPTS * 512 * 4);   // x1|x2|x3|x4 slices
  float* x5   = (float*)alloc((size_t)BB * NPTS * 1024 * 4);
  float* d2b  = (float*)alloc((size_t)BB * NPTS * 4);
  int*   idxb = (int*)  alloc((size_t)BB * NPTS * KNN * 4);
  float* pmax = (float*)alloc((size_t)BB * 16 * 1024 * 4);
  float* psum = (float*)alloc((size_t)BB * 16 * 1024 * 4);
  float* g    = (float*)alloc((size_t)BB * 2048 * 4);
  float* h1   = (float*)alloc((size_t)BB * 512 * 4);
  float* h2   = (float*)alloc((size_t)BB * 256 * 4);

  const int PB = BB * NPTS / 16;            // 1024 tile-blocks
  const int TB = (BB * NPTS + 255) / 256;

  // ---- EdgeConv 1: x[.,3] -> xc[:, 0:64]
  d2_kernel<<<TB, 256, 0, stream>>>(x, 3, 0, 3, d2b);
  knn_kernel<3><<<PB, 256, 0, stream>>>(x, 3, 0, d2b, idxb);
  edgeconv_kernel<3, 64, 4><<<PB, 128, 0, stream>>>(x, 3, 0, idxb, w1, bn1, xc, 512, 0);
  // ---- EdgeConv 2: xc[:,0:64] -> xc[:, 64:128]
  d2_kernel<<<TB, 256, 0, stream>>>(xc, 512, 0, 64, d2b);
  knn_kernel<64><<<PB, 256, 0, stream>>>(xc, 512, 0, d2b, idxb);
  edgeconv_kernel<64, 64, 4><<<PB, 128, 0, stream>>>(xc, 512, 0, idxb, w2, bn2, xc, 512, 64);
  // ---- EdgeConv 3: xc[:,64:128] -> xc[:, 128:256]
  d2_kernel<<<TB, 256, 0, stream>>>(xc, 512, 64, 64, d2b);
  knn_kernel<64><<<PB, 256, 0, stream>>>(xc, 512, 64, d2b, idxb);
  edgeconv_kernel<64, 128, 8><<<PB, 256, 0, stream>>>(xc, 512, 64, idxb, w3, bn3, xc, 512, 128);
  // ---- EdgeConv 4: xc[:,128:256] -> xc[:, 256:512]
  d2_kernel<<<TB, 256, 0, stream>>>(xc, 512, 128, 128, d2b);
  knn_kernel<128><<<PB, 256, 0, stream>>>(xc, 512, 128, d2b, idxb);
  edgeconv_kernel<128, 256, 8><<<PB, 256, 0, stream>>>(xc, 512, 128, idxb, w4, bn4, xc, 512, 256);

  // ---- x5 = act(bn5(xc @ w5^T)): [B*N,512] x [1024,512]^T
  rowgemm_bn_act_kernel<512, 1024, 8><<<PB, 256, 0, stream>>>(xc, 512, 0, w5, bn5, x5, 1024);

  // ---- global max+mean pooling -> g[B,2048]
  pool1_kernel<<<dim3(16, BB), 256, 0, stream>>>(x5, pmax, psum);
  pool2_kernel<<<(BB * 1024 + 255) / 256, 256, 0, stream>>>(pmax, psum, g);

  // ---- FC head
  head_fc_kernel<<<(BB * 512 + 255) / 256, 256, 0, stream>>>(g,  wl1, nullptr, bn6, h1, 2048, 512, 1);
  head_fc_kernel<<<(BB * 256 + 255) / 256, 256, 0, stream>>>(h1, wl2, bl2,     bn7, h2, 512,  256, 1);
  head_fc_kernel<<<(BB * 40  + 255) / 256, 256, 0, stream>>>(h2, wl3, bl3, nullptr, out, 256, 40, 0);
}